// LSM_IniReconNet_23278722744926
// MI455X (gfx1250) — compile-verified
//
#include <hip/hip_runtime.h>

typedef __attribute__((ext_vector_type(2))) float v2f;
typedef __attribute__((ext_vector_type(8))) float v8f;

#define SP 8
#define BLK 16

// Kernel 1: fold the two tiny weight matrices into one 16x16 matrix M.
// M[k][n] = sum_s W1[s][k] * W2[n][s]
//   W1[s][k] = W_samp[s*16 + k]   (W_samp is (8,1,16))
//   W2[n][s] = W_init[n*8  + s]   (W_init is (16,8,1))
__global__ void build_M_kernel(const float* __restrict__ W_samp,
                               const float* __restrict__ W_init,
                               float* __restrict__ M) {
    const int t = threadIdx.x;          // 0..255
    const int k = t >> 4;
    const int n = t & 15;
    float acc = 0.0f;
#pragma unroll
    for (int s = 0; s < SP; ++s)
        acc += W_samp[s * BLK + k] * W_init[n * SP + s];
    M[k * BLK + n] = acc;
}

// Kernel 2: stream rows of x, apply y_tile = A_tile(16x16) @ M(16x16) via
// four chained V_WMMA_F32_16X16X4_F32 ops per tile. One wave per 16-row tile.
__global__ void __launch_bounds__(256) apply_M_kernel(
        const float* __restrict__ x,
        const float* __restrict__ M,
        float* __restrict__ y,
        int ntiles) {
    const int lane   = threadIdx.x & 31;
    const int wib    = threadIdx.x >> 5;
    const int gwave  = blockIdx.x * (blockDim.x >> 5) + wib;
    const int nwaves = gridDim.x * (blockDim.x >> 5);
    const int h = lane >> 4;            // half-wave select (K offset / row offset)
    const int n = lane & 15;            // column / row-in-tile index

    // B operand (M), ISA 4x16 layout: VGPR0 = K=4c+2h, VGPR1 = K=4c+2h+1,
    // N = lane&15. Loaded once, reused for every tile.
    v2f b[4];
#pragma unroll
    for (int c = 0; c < 4; ++c) {
        b[c].x = M[(4 * c + 2 * h + 0) * BLK + n];
        b[c].y = M[(4 * c + 2 * h + 1) * BLK + n];
    }

    for (int tile = gwave; tile < ntiles; tile += nwaves) {
        const float* Arow = x + (size_t)tile * 256 + (size_t)n * BLK;

        // Prefetch this lane's row of the next tile this wave will process.
        __builtin_prefetch(Arow + (size_t)nwaves * 256, 0, 0);

        // A operand (16x4 f32 layout): lane n = row, chunk c covers K=4c..4c+3,
        // this lane holds the contiguous pair (4c+2h, 4c+2h+1) -> b64 loads.
        v2f a[4];
#pragma unroll
        for (int c = 0; c < 4; ++c)
            a[c] = *(const v2f*)(Arow + 4 * c + 2 * h);

        v8f d = {};
#pragma unroll
        for (int c = 0; c < 4; ++c)
            d = __builtin_amdgcn_wmma_f32_16x16x4_f32(
                    /*neg_a=*/false, a[c],
                    /*neg_b=*/false, b[c],
                    /*c_mod=*/(short)0, d,
                    /*reuse_a=*/false, /*reuse_b=*/false);

        // C/D layout: d[g] -> row g + 8h, col n. 8 coalesced 64B row stores.
        float* Drow = y + (size_t)tile * 256;
#pragma unroll
        for (int g = 0; g < 8; ++g)
            Drow[(size_t)(g + 8 * h) * BLK + n] = d[g];
    }
}

extern "C" void kernel_launch(void* const* d_in, const int* in_sizes, int n_in,
                              void* d_out, int out_size, void* d_ws, size_t ws_size,
                              hipStream_t stream) {
    const float* x      = (const float*)d_in[0];   // (4096,1,4096,1) fp32
    const float* W_samp = (const float*)d_in[1];   // (8,1,16) fp32
    const float* W_init = (const float*)d_in[2];   // (16,8,1) fp32
    float* y = (float*)d_out;                      // (4096,4096,1) fp32
    float* M = (float*)d_ws;                       // 256 floats scratch

    // Stage 1: fold weights into 16x16 M (stream-ordered before stage 2).
    build_M_kernel<<<1, 256, 0, stream>>>(W_samp, W_init, M);

    // Stage 2: memory-bound streaming WMMA apply.
    const int ntiles = in_sizes[0] / 256;          // 65536 tiles of 16 rows
    const int blocks = 1024;                       // 8192 waves, 8 tiles each
    apply_M_kernel<<<blocks, 256, 0, stream>>>(x, M, y, ntiles);
}